// BatchTtripletLoss_42099269435804
// MI455X (gfx1250) — compile-verified
//
#include <hip/hip_runtime.h>
#include <hip/hip_bf16.h>

#define B 512
#define E 128
#define MARGIN 1.0f
#define EPS 1e-8f

typedef __attribute__((ext_vector_type(2))) float v2f;
typedef __attribute__((ext_vector_type(8))) float v8f;

// ---------------------------------------------------------------------------
// Kernel 0: per-row squared norms  sq[i] = sum_k X[i][k]^2
// ---------------------------------------------------------------------------
__global__ void rownorm_kernel(const float* __restrict__ X, float* __restrict__ sq) {
    int i = blockIdx.x * blockDim.x + threadIdx.x;
    if (i < B) {
        const float* row = X + i * E;
        float s = 0.f;
#pragma unroll 8
        for (int k = 0; k < E; ++k) {
            float v = row[k];
            s += v * v;
        }
        sq[i] = s;
    }
}

// ---------------------------------------------------------------------------
// Kernel 1: Gram tile via V_WMMA_F32_16X16X4_F32, fused into distance matrix.
// One wave (32 lanes) per 16x16 tile. Gram = X * X^T so both A (16x4) and
// B (4x16) tiles are rows of X with the same per-lane addressing:
//   lane L, VGPR r  ->  k = kbase + 2*(L>>4) + r   (ISA 7.12.2 layouts)
// C/D layout: VGPR r, lane L -> M = r + 8*(L>>4), N = L&15.
// ---------------------------------------------------------------------------
__global__ void gram_dist_kernel(const float* __restrict__ X,
                                 const float* __restrict__ sq,
                                 float* __restrict__ dist) {
    const int tile_m = blockIdx.x * 16;
    const int tile_n = blockIdx.y * 16;
    const int lane  = threadIdx.x;        // 0..31, full wave32, EXEC all 1s
    const int mn    = lane & 15;
    const int half  = lane >> 4;

    const float* arow = X + (tile_m + mn) * E + 2 * half;   // A: rows of X
    const float* brow = X + (tile_n + mn) * E + 2 * half;   // B: rows of X (transposed use)

    v8f acc = {};
#pragma unroll 4
    for (int kb = 0; kb < E; kb += 4) {
        v2f a = *reinterpret_cast<const v2f*>(arow + kb);   // 8B-aligned
        v2f b = *reinterpret_cast<const v2f*>(brow + kb);
        acc = __builtin_amdgcn_wmma_f32_16x16x4_f32(
            /*neg_a=*/false, a, /*neg_b=*/false, b,
            /*c_mod=*/(short)0, acc, /*reuse_a=*/false, /*reuse_b=*/false);
    }

    // Fused distance transform: d2 = sq[i] - 2*dot + sq[j]; relu; zero-mask sqrt
    const float sqn = sq[tile_n + mn];
#pragma unroll
    for (int r = 0; r < 8; ++r) {
        const int M = tile_m + r + 8 * half;
        float d2 = sq[M] - 2.0f * acc[r] + sqn;
        d2 = d2 > 0.0f ? d2 : 0.0f;
        const float zm = (d2 == 0.0f) ? 1.0f : 0.0f;
        const float d  = sqrtf(d2 + zm * EPS) * (1.0f - zm);
        dist[M * B + tile_n + mn] = d;
    }
}

// ---------------------------------------------------------------------------
// Kernel 2: triplet reduction. One block per anchor i; dist row + labels in
// LDS; uniform j scan (positives ~ B/C), lane-parallel k scan (negatives).
// Writes per-anchor partial (sum, count) -> deterministic, no atomics.
// ---------------------------------------------------------------------------
__global__ void triplet_kernel(const float* __restrict__ dist,
                               const int* __restrict__ labels,
                               float* __restrict__ psum,
                               float* __restrict__ pcnt) {
    const int i = blockIdx.x;
    const int t = threadIdx.x;            // blockDim.x == 256

    __shared__ float drow[B];
    __shared__ int   slbl[B];
    __shared__ float ssum[256];
    __shared__ float scnt[256];

    for (int c = t; c < B; c += 256) {
        drow[c] = dist[i * B + c];
        slbl[c] = labels[c];
    }
    __syncthreads();

    const int li = slbl[i];
    float lsum = 0.f, lcnt = 0.f;

    for (int j = 0; j < B; ++j) {
        if (j == i) continue;
        if (slbl[j] != li) continue;      // uniform across block
        const float dij = drow[j] + MARGIN;
#pragma unroll
        for (int k = t; k < B; k += 256) {
            if (slbl[k] == li) continue;  // also excludes k==i, k==j
            const float v = dij - drow[k];
            if (v > 0.0f) {
                lsum += v;
                if (v > EPS) lcnt += 1.0f;
            }
        }
    }

    ssum[t] = lsum;
    scnt[t] = lcnt;
    __syncthreads();
    for (int off = 128; off > 0; off >>= 1) {
        if (t < off) { ssum[t] += ssum[t + off]; scnt[t] += scnt[t + off]; }
        __syncthreads();
    }
    if (t == 0) { psum[i] = ssum[0]; pcnt[i] = scnt[0]; }
}

// ---------------------------------------------------------------------------
// Kernel 3: reduce 512 partials -> out = sum / (count + eps)
// ---------------------------------------------------------------------------
__global__ void finalize_kernel(const float* __restrict__ psum,
                                const float* __restrict__ pcnt,
                                float* __restrict__ out) {
    const int t = threadIdx.x;            // 256 threads
    __shared__ float ss[256];
    __shared__ float sc[256];
    ss[t] = psum[t] + psum[t + 256];
    sc[t] = pcnt[t] + pcnt[t + 256];
    __syncthreads();
    for (int off = 128; off > 0; off >>= 1) {
        if (t < off) { ss[t] += ss[t + off]; sc[t] += sc[t + off]; }
        __syncthreads();
    }
    if (t == 0) out[0] = ss[0] / (sc[0] + EPS);
}

// ---------------------------------------------------------------------------
extern "C" void kernel_launch(void* const* d_in, const int* in_sizes, int n_in,
                              void* d_out, int out_size, void* d_ws, size_t ws_size,
                              hipStream_t stream) {
    const float* X      = (const float*)d_in[0];   // (512, 128) fp32
    const int*   labels = (const int*)d_in[1];     // (512,) int32 (JAX x64 off)
    float*       out    = (float*)d_out;           // scalar fp32

    char*  ws   = (char*)d_ws;
    float* dist = (float*)ws;                      // 512*512 floats = 1 MB
    float* sq   = (float*)(ws + (size_t)B * B * sizeof(float));  // 512
    float* psum = sq + B;                          // 512
    float* pcnt = psum + B;                        // 512

    rownorm_kernel<<<(B + 255) / 256, 256, 0, stream>>>(X, sq);
    gram_dist_kernel<<<dim3(B / 16, B / 16), 32, 0, stream>>>(X, sq, dist);
    triplet_kernel<<<B, 256, 0, stream>>>(dist, labels, psum, pcnt);
    finalize_kernel<<<1, 256, 0, stream>>>(psum, pcnt, out);
}